// MetaWaveletProjector_89300960018581
// MI455X (gfx1250) — compile-verified
//
#include <hip/hip_runtime.h>
#include <cmath>
#include <stdint.h>

typedef __attribute__((ext_vector_type(2))) float v2f;
typedef __attribute__((ext_vector_type(8))) float v8f;

#define N_ROWS 16384
#define D_IN   128
#define CTXK   256
#define HID    512
#define PDIM   512
#define NF     32
#define FL     16
#define LOUT   113
#define OUT_STRIDE (NF * LOUT) /* 3616 */

__device__ __forceinline__ float gelu_exact(float v) {
    // exact GELU: 0.5*x*(1+erf(x/sqrt(2)))
    return 0.5f * v * (1.0f + erff(v * 0.70710678118654752f));
}

__device__ __forceinline__ v8f wmma4(v2f a, v2f b, v8f c) {
    // D = A(16x4 f32) * B(4x16 f32) + C(16x16 f32), full fp32 precision
    return __builtin_amdgcn_wmma_f32_16x16x4_f32(
        /*neg_a=*/false, a, /*neg_b=*/false, b,
        /*c_mod=*/(short)0, c, /*reuse_a=*/false, /*reuse_b=*/false);
}

// ---------------------------------------------------------------------------
// Kernel 1: h = gelu(context @ W1 + b1)   [16384 x 256] @ [256 x 512]
// Workgroup (8 waves) covers a 64-row x 128-col block (4x2 wave grid of
// 16x64 per-wave tiles) to shrink the per-WG weight footprint to K x 128 and
// reuse A rows across the 2 column-group waves through L1.
// grid = (16384/64) * (512/128) = 1024 blocks.
// ---------------------------------------------------------------------------
__global__ __launch_bounds__(256) void k_gemm1_gelu(
    const float* __restrict__ A, const float* __restrict__ W,
    const float* __restrict__ bias, float* __restrict__ H)
{
    const int lane = threadIdx.x & 31;
    const int wave = threadIdx.x >> 5;
    const int m    = lane & 15;
    const int half = lane >> 4;
    const int klo  = half * 2;
    const int rowb = (blockIdx.x >> 2) * 64 + (wave >> 1) * 16;
    const int colb = (blockIdx.x & 3) * 128 + (wave & 1) * 64;

    v8f c0 = {}, c1 = {}, c2 = {}, c3 = {};
    const float* arow = A + (size_t)(rowb + m) * CTXK + klo;

    #pragma unroll 4
    for (int k0 = 0; k0 < CTXK; k0 += 4) {
        v2f a = *(const v2f*)(arow + k0);                    // A[m][k0+klo .. +1]
        const float* bp = W + (size_t)(k0 + klo) * HID + colb + m;
        v2f b0 = { bp[0],  bp[0  + HID] };                   // B[k][n], B[k+1][n]
        v2f b1 = { bp[16], bp[16 + HID] };
        v2f b2 = { bp[32], bp[32 + HID] };
        v2f b3 = { bp[48], bp[48 + HID] };
        c0 = wmma4(a, b0, c0);
        c1 = wmma4(a, b1, c1);
        c2 = wmma4(a, b2, c2);
        c3 = wmma4(a, b3, c3);
    }

    v8f acc[4] = { c0, c1, c2, c3 };
    #pragma unroll
    for (int t = 0; t < 4; ++t) {
        const int col = colb + t * 16 + m;
        const float bv = bias[col];
        #pragma unroll
        for (int i = 0; i < 8; ++i) {
            const int row = rowb + half * 8 + i;
            H[(size_t)row * HID + col] = gelu_exact(acc[t][i] + bv);
        }
    }
}

// ---------------------------------------------------------------------------
// Kernel 2: theta = h @ W2 + b2, then per-16-element filter L2-normalize.
// A 16x16 C-tile spans exactly one filter group per row-half, so the norm is
// a 16-lane xor-shuffle reduction matching the WMMA C/D layout.
// ---------------------------------------------------------------------------
__global__ __launch_bounds__(256) void k_gemm2_norm(
    const float* __restrict__ A, const float* __restrict__ W,
    const float* __restrict__ bias, float* __restrict__ F)
{
    const int lane = threadIdx.x & 31;
    const int wave = threadIdx.x >> 5;
    const int m    = lane & 15;
    const int half = lane >> 4;
    const int klo  = half * 2;
    const int rowb = (blockIdx.x >> 2) * 64 + (wave >> 1) * 16;
    const int colb = (blockIdx.x & 3) * 128 + (wave & 1) * 64;

    v8f c0 = {}, c1 = {}, c2 = {}, c3 = {};
    const float* arow = A + (size_t)(rowb + m) * HID + klo;

    #pragma unroll 4
    for (int k0 = 0; k0 < HID; k0 += 4) {
        v2f a = *(const v2f*)(arow + k0);
        const float* bp = W + (size_t)(k0 + klo) * PDIM + colb + m;
        v2f b0 = { bp[0],  bp[0  + PDIM] };
        v2f b1 = { bp[16], bp[16 + PDIM] };
        v2f b2 = { bp[32], bp[32 + PDIM] };
        v2f b3 = { bp[48], bp[48 + PDIM] };
        c0 = wmma4(a, b0, c0);
        c1 = wmma4(a, b1, c1);
        c2 = wmma4(a, b2, c2);
        c3 = wmma4(a, b3, c3);
    }

    v8f acc[4] = { c0, c1, c2, c3 };
    #pragma unroll
    for (int t = 0; t < 4; ++t) {
        const int col = colb + t * 16 + m;
        const float bv = bias[col];
        #pragma unroll
        for (int i = 0; i < 8; ++i) {
            const int row = rowb + half * 8 + i;
            float v = acc[t][i] + bv;
            // sum of squares over the 16 columns of this filter (16-lane half)
            float s = v * v;
            s += __shfl_xor(s, 1, 32);
            s += __shfl_xor(s, 2, 32);
            s += __shfl_xor(s, 4, 32);
            s += __shfl_xor(s, 8, 32);
            const float sc = 1.0f / (sqrtf(s) + 1e-8f);
            F[(size_t)row * PDIM + col] = v * sc;
        }
    }
}

// ---------------------------------------------------------------------------
// Kernel 3: coeffs[n,f,w] = sum_l x[n,w+l] * filt[n,f,l]
// Per row: D(32x128 padded) = filters(32x16) x windows^T(16x128) via f32 WMMA.
// One workgroup (8 waves) per row; wave = one 16-wide w tile, two f tiles.
// LDS staging uses gfx1250 async memory->LDS loads (ASYNCcnt), bypassing VGPRs.
// ---------------------------------------------------------------------------
__global__ __launch_bounds__(256) void k_coeffs(
    const float* __restrict__ X, const float* __restrict__ F,
    float* __restrict__ OUT)
{
    __shared__ float xs[160];   // 128 + zero pad (max read index 142)
    __shared__ float fs[PDIM];  // 32 filters x 16 taps

    const int n = blockIdx.x;
    const int t = threadIdx.x;

    // zero the window padding region (w >= 113 tiles read up to xs[142])
    if (t >= 128 && t < 160) xs[t] = 0.0f;

    // Async stage filters: 256 threads x b64  (512 floats)
    {
        const float* gsrc = F + (size_t)n * PDIM + t * 2;
        uint32_t lds_f = (uint32_t)(uintptr_t)(fs + t * 2);
        asm volatile("global_load_async_to_lds_b64 %0, %1, off"
                     :: "v"(lds_f), "v"(gsrc) : "memory");
    }
    // Async stage x row: 128 threads x b32
    if (t < D_IN) {
        const float* gx = X + (size_t)n * D_IN + t;
        uint32_t lds_x = (uint32_t)(uintptr_t)(xs + t);
        asm volatile("global_load_async_to_lds_b32 %0, %1, off"
                     :: "v"(lds_x), "v"(gx) : "memory");
    }
    asm volatile("s_wait_asynccnt 0x0" ::: "memory");
    __syncthreads();

    const int lane = t & 31;
    const int wave = t >> 5;
    const int m    = lane & 15;
    const int half = lane >> 4;
    const int klo  = half * 2;
    const int w    = wave * 16 + m;   // output window position (padded to 128)

    v8f c0 = {}, c1 = {};
    #pragma unroll
    for (int k0 = 0; k0 < FL; k0 += 4) {
        const int kk = k0 + klo;
        v2f b  = { xs[w + kk], xs[w + kk + 1] };            // windows^T[k][w]
        v2f a0 = *(const v2f*)(fs + m * FL + kk);           // filter f = m
        v2f a1 = *(const v2f*)(fs + (16 + m) * FL + kk);    // filter f = 16+m
        c0 = wmma4(a0, b, c0);
        c1 = wmma4(a1, b, c1);
    }

    if (w < LOUT) {
        float* orow = OUT + (size_t)n * OUT_STRIDE + w;
        #pragma unroll
        for (int i = 0; i < 8; ++i) {
            const int f0 = half * 8 + i;                    // f in [0,16)
            orow[f0 * LOUT]        = c0[i];
            orow[(16 + f0) * LOUT] = c1[i];
        }
    }
}

// ---------------------------------------------------------------------------
extern "C" void kernel_launch(void* const* d_in, const int* in_sizes, int n_in,
                              void* d_out, int out_size, void* d_ws, size_t ws_size,
                              hipStream_t stream) {
    (void)in_sizes; (void)n_in; (void)out_size; (void)ws_size;

    const float* x   = (const float*)d_in[0];  // (8,2048,128)
    const float* ctx = (const float*)d_in[1];  // (8,2048,256)
    const float* W1  = (const float*)d_in[2];  // (256,512)
    const float* b1  = (const float*)d_in[3];  // (512)
    const float* W2  = (const float*)d_in[4];  // (512,512)
    const float* b2  = (const float*)d_in[5];  // (512)
    float* out = (float*)d_out;                // (8,2048,3616)

    float* h    = (float*)d_ws;                      // 16384*512 f32 = 33.5 MB
    float* filt = h + (size_t)N_ROWS * HID;          // 16384*512 f32 = 33.5 MB

    k_gemm1_gelu<<<(N_ROWS / 64) * (HID  / 128), 256, 0, stream>>>(ctx, W1, b1, h);
    k_gemm2_norm<<<(N_ROWS / 64) * (PDIM / 128), 256, 0, stream>>>(h, W2, b2, filt);
    k_coeffs    <<<N_ROWS,                       256, 0, stream>>>(x, filt, out);
}